// BlockDense_15393162788948
// MI455X (gfx1250) — compile-verified
//
#include <hip/hip_runtime.h>

typedef __attribute__((ext_vector_type(2))) float v2f;
typedef __attribute__((ext_vector_type(8))) float v8f;

#define G_CNT   1024
#define K_SZ    16      // W_SZ
#define H_SZ    16
#define B_ROWS  8192
#define ROW_F   (G_CNT * K_SZ)   // 16384 floats per x row
#define OROW_F  (G_CNT * H_SZ)   // 16384 floats per out row
#define TILES_PER_WAVE 8

// One wave (32 lanes) computes a 16x16 fp32 output tile for one group g:
//   out[b0:b0+16, g*16:(g+1)*16] = relu( x[b0:b0+16, g*16:(g+1)*16] @ W[g] )
// via 4 chained V_WMMA_F32_16X16X4_F32 over K=16.
__global__ __launch_bounds__(256) void blockdense_wmma_f32(
    const float* __restrict__ x,
    const float* __restrict__ Wt,
    float* __restrict__ out)
{
    const int lane  = threadIdx.x & 31;
    const int wave  = threadIdx.x >> 5;
    const int g     = ((blockIdx.x & 127) << 3) + wave;   // 0..1023
    const int chunk = blockIdx.x >> 7;                    // 0..63
    const int m     = lane & 15;                          // row-in-tile / col-in-tile
    const int khalf = lane >> 4;                          // 0: K pair {0,1}; 1: K pair {2,3}

    // ---- B fragments (weights), hoisted: 4 v2f frags, reused for all batch tiles.
    // ISA B-layout (4x16 f32): lanes 0-15 hold B[k0+0][n], B[k0+1][n];
    //                          lanes 16-31 hold B[k0+2][n], B[k0+3][n].
    const float* wg = Wt + (size_t)g * (K_SZ * H_SZ);
    v2f bfrag[4];
#pragma unroll
    for (int kk = 0; kk < 4; ++kk) {
        const int krow = kk * 4 + khalf * 2;
        v2f b;
        b.x = wg[(krow + 0) * H_SZ + m];
        b.y = wg[(krow + 1) * H_SZ + m];
        bfrag[kk] = b;
    }

    const int row0 = chunk * (TILES_PER_WAVE * 16);

#pragma unroll 1
    for (int it = 0; it < TILES_PER_WAVE; ++it) {
        const int b0 = row0 + it * 16;
        // ISA A-layout (16x4 f32): lanes 0-15: A[m][k0+0..1]; lanes 16-31: A[m][k0+2..3]
        const float* xrow = x + (size_t)(b0 + m) * ROW_F + g * K_SZ + khalf * 2;

        // Prefetch next tile's row segment (emits global_prefetch_b8).
        if (it + 1 < TILES_PER_WAVE)
            __builtin_prefetch(xrow + (size_t)16 * ROW_F, 0, 1);

        v8f acc = {};   // zero-init C
#pragma unroll
        for (int kk = 0; kk < 4; ++kk) {
            v2f a = *(const v2f*)(xrow + kk * 4);   // 8B aligned -> global_load_b64
            acc = __builtin_amdgcn_wmma_f32_16x16x4_f32(
                /*neg_a=*/false, a, /*neg_b=*/false, bfrag[kk],
                /*c_mod=*/(short)0, acc, /*reuse_a=*/false, /*reuse_b=*/false);
        }

        // D layout: VGPR r -> out row b0 + r + (khalf ? 8 : 0), col g*16 + m.
        float* orow = out + (size_t)(b0 + khalf * 8) * OROW_F + g * H_SZ + m;
#pragma unroll
        for (int r = 0; r < 8; ++r) {
            float v = acc[r];
            v = v > 0.0f ? v : 0.0f;               // ReLU
            orow[(size_t)r * OROW_F] = v;
        }
    }
}

extern "C" void kernel_launch(void* const* d_in, const int* in_sizes, int n_in,
                              void* d_out, int out_size, void* d_ws, size_t ws_size,
                              hipStream_t stream) {
    (void)in_sizes; (void)n_in; (void)out_size; (void)d_ws; (void)ws_size;
    const float* x  = (const float*)d_in[0];   // (8192, 16384) f32
    const float* Wt = (const float*)d_in[1];   // (1024, 16, 16) f32
    float* out      = (float*)d_out;           // (8192, 16384) f32

    // 128 group-columns (8 groups/block, one per wave) x 64 batch chunks (128 rows each)
    const int nblocks = 128 * (B_ROWS / (TILES_PER_WAVE * 16));  // 8192
    blockdense_wmma_f32<<<nblocks, 256, 0, stream>>>(x, Wt, out);
}